// DecoderWithAttention_51342039056422
// MI455X (gfx1250) — compile-verified
//
#include <hip/hip_runtime.h>
#include <math.h>

#define B_   256
#define P_   100
#define ENC_ 1536
#define DEC_ 512
#define ATT_ 256
#define EMB_ 256
#define V_   193
#define T_   150

typedef __attribute__((ext_vector_type(2))) float v2f;
typedef __attribute__((ext_vector_type(8))) float v8f;

__device__ __forceinline__ float sigmoidf_(float x) { return 1.0f / (1.0f + expf(-x)); }

// ---------------------------------------------------------------------------
// K-loop helper: accumulate A(64 rows x K) @ W(K x 16cols) into 4 WMMA tiles.
// mrow = first of 64 A rows for this lane's l; ncol = clamped B column.
// Per k-step: 4 aligned v2f (b64) A loads, 2 B loads, 4 V_WMMA_F32_16X16X4_F32.
// ---------------------------------------------------------------------------
__device__ __forceinline__ void gemm_kloop(v8f acc[4],
                                           const float* __restrict__ A, int lda,
                                           const float* __restrict__ W, int ldw, int K,
                                           int mrow, int ncol, int half)
{
    const size_t ldwz = (size_t)ldw;
    const float* __restrict__ a0 = A + (size_t)(mrow +  0) * lda + 2 * half;
    const float* __restrict__ a1 = A + (size_t)(mrow + 16) * lda + 2 * half;
    const float* __restrict__ a2 = A + (size_t)(mrow + 32) * lda + 2 * half;
    const float* __restrict__ a3 = A + (size_t)(mrow + 48) * lda + 2 * half;
    const float* __restrict__ w  = W + (size_t)(2 * half) * ldwz + ncol;

    for (int k = 0; k < K; k += 4) {
        v2f b;
        b.x = w[0];
        b.y = w[ldwz];
        w += 4 * ldwz;

        v2f av0 = *(const v2f*)a0; a0 += 4;
        v2f av1 = *(const v2f*)a1; a1 += 4;
        v2f av2 = *(const v2f*)a2; a2 += 4;
        v2f av3 = *(const v2f*)a3; a3 += 4;

        acc[0] = __builtin_amdgcn_wmma_f32_16x16x4_f32(false, av0, false, b, (short)0, acc[0], false, false);
        acc[1] = __builtin_amdgcn_wmma_f32_16x16x4_f32(false, av1, false, b, (short)0, acc[1], false, false);
        acc[2] = __builtin_amdgcn_wmma_f32_16x16x4_f32(false, av2, false, b, (short)0, acc[2], false, false);
        acc[3] = __builtin_amdgcn_wmma_f32_16x16x4_f32(false, av3, false, b, (short)0, acc[3], false, false);
    }
}

// ---------------------------------------------------------------------------
// Fused fp32 WMMA GEMM:
//   C(M,N) = A1@W1 [+ A2@W2] [+ A3@W3] [+ bias]
// Each wave owns a 64x16 C tile. Requires M % 64 == 0 (true at all call
// sites) and ldw == ldc == N for all weights (true at all call sites).
// mode 0: plain   mode 1: C = aux * sigmoid(v)   mode 2: C = (t<lens[row]-1)?v:0
// N may be ragged (V=193): B column is clamped, stores are guarded.
// ---------------------------------------------------------------------------
__global__ void wmma_gemm_f32(const float* __restrict__ A1, int lda1, const float* __restrict__ W1, int K1,
                              const float* __restrict__ A2, int lda2, const float* __restrict__ W2, int K2,
                              const float* __restrict__ A3, int lda3, const float* __restrict__ W3, int K3,
                              const float* __restrict__ bias,
                              float* __restrict__ C, int M, int N,
                              int mode,
                              const float* __restrict__ aux, int ldaux,
                              const int* __restrict__ lens, int tstep)
{
    const int lane = threadIdx.x & 31;
    const int half = lane >> 4;        // 0 or 1 (K-split halves)
    const int l    = lane & 15;        // 0..15
    const int m0   = blockIdx.y * 64;
    const int n0   = (blockIdx.x * blockDim.y + threadIdx.y) * 16;
    if (n0 >= N) return;               // wave-uniform exit, EXEC stays all-ones

    const int ncol = min(n0 + l, N - 1);   // clamped: loads always in-bounds
    const int mrow = m0 + l;

    v8f acc[4] = {};

    gemm_kloop(acc, A1, lda1, W1, N, K1, mrow, ncol, half);
    if (A2) gemm_kloop(acc, A2, lda2, W2, N, K2, mrow, ncol, half);
    if (A3) gemm_kloop(acc, A3, lda3, W3, N, K3, mrow, ncol, half);

    const int col = n0 + l;
    const bool colOk = col < N;
    float bv = 0.0f;
    if (bias) bv = bias[ncol];

#pragma unroll
    for (int i = 0; i < 4; ++i) {
#pragma unroll
        for (int r = 0; r < 8; ++r) {
            int row = m0 + 16 * i + r + 8 * half;
            if (colOk) {
                float v = acc[i][r] + bv;
                if (mode == 1)      v = aux[(size_t)row * ldaux + col] * sigmoidf_(v);
                else if (mode == 2) v = (tstep < (lens[row] - 1)) ? v : 0.0f;
                C[(size_t)row * N + col] = v;
            }
        }
    }
}

// ---------------------------------------------------------------------------
// Attention core: e = relu(att1[b] + att2[b]) @ W_full + b_full ; softmax over P ;
// awe[b,:] = alpha @ encoder_out[b]. One block (8 waves) per batch row.
// ---------------------------------------------------------------------------
__global__ void attention_kernel(const float* __restrict__ att1,
                                 const float* __restrict__ att2,
                                 const float* __restrict__ Wfull,
                                 const float* __restrict__ bfull,
                                 const float* __restrict__ enc,
                                 float* __restrict__ awe)
{
    __shared__ float s_att2[ATT_];
    __shared__ float s_e[128];

    const int b    = blockIdx.x;
    const int tid  = threadIdx.x;      // 0..255
    const int lane = tid & 31;
    const int wid  = tid >> 5;

    s_att2[tid] = att2[b * ATT_ + tid];
    __syncthreads();

    for (int p = wid; p < P_; p += 8) {
        const float* __restrict__ row = att1 + (size_t)(b * P_ + p) * ATT_;
        float sum = 0.0f;
#pragma unroll
        for (int a = lane; a < ATT_; a += 32) {
            float v = fmaxf(row[a] + s_att2[a], 0.0f);
            sum = fmaf(v, Wfull[a], sum);
        }
#pragma unroll
        for (int off = 16; off > 0; off >>= 1) sum += __shfl_xor(sum, off, 32);
        if (lane == 0) s_e[p] = sum + bfull[0];
    }
    __syncthreads();

    if (wid == 0) {
        float mx = -3.4e38f;
        for (int p = lane; p < P_; p += 32) mx = fmaxf(mx, s_e[p]);
#pragma unroll
        for (int off = 16; off > 0; off >>= 1) mx = fmaxf(mx, __shfl_xor(mx, off, 32));
        float sm = 0.0f;
        for (int p = lane; p < P_; p += 32) { float ex = expf(s_e[p] - mx); s_e[p] = ex; sm += ex; }
#pragma unroll
        for (int off = 16; off > 0; off >>= 1) sm += __shfl_xor(sm, off, 32);
        float inv = 1.0f / sm;
        for (int p = lane; p < P_; p += 32) s_e[p] *= inv;
    }
    __syncthreads();

    for (int e = tid; e < ENC_; e += 256) {
        const float* __restrict__ ebase = enc + (size_t)b * P_ * ENC_ + e;
        float acc = 0.0f;
#pragma unroll 4
        for (int p = 0; p < P_; ++p) acc = fmaf(s_e[p], ebase[(size_t)p * ENC_], acc);
        awe[b * ENC_ + e] = acc;
    }
}

// mean over the P axis of encoder_out: mean[b, e] = (1/P) sum_p enc[b,p,e]
__global__ void mean_kernel(const float* __restrict__ enc, float* __restrict__ mean)
{
    const int b = blockIdx.x;
    for (int e = threadIdx.x; e < ENC_; e += blockDim.x) {
        float s = 0.0f;
        for (int p = 0; p < P_; ++p) s += enc[(size_t)(b * P_ + p) * ENC_ + e];
        mean[b * ENC_ + e] = s * (1.0f / (float)P_);
    }
}

// per-step embedding gather: embt[b, :] = emb_table[captions[b, t]]
__global__ void emb_gather(const int* __restrict__ caps, const float* __restrict__ table,
                           float* __restrict__ embt, int t)
{
    int i = blockIdx.x * blockDim.x + threadIdx.x;
    if (i >= B_ * EMB_) return;
    int b = i / EMB_, e = i - b * EMB_;
    int tok = caps[b * (T_ + 1) + t];
    embt[i] = table[tok * EMB_ + e];
}

// LSTM cell elementwise with length mask. The fused gates GEMM applied only
// b_ih; the reference adds b_ih + b_hh, so b_hh is added here per gate.
__global__ void lstm_cell(const float* __restrict__ gates,
                          const float* __restrict__ b_hh,
                          float* __restrict__ h, float* __restrict__ c,
                          const int* __restrict__ lens, int t)
{
    int i = blockIdx.x * blockDim.x + threadIdx.x;
    if (i >= B_ * DEC_) return;
    int b = i / DEC_, d = i - b * DEC_;
    const float* __restrict__ g = gates + (size_t)b * 4 * DEC_;
    float ig = sigmoidf_(g[d]            + b_hh[d]);
    float fg = sigmoidf_(g[DEC_ + d]     + b_hh[DEC_ + d]);
    float gg = tanhf(    g[2 * DEC_ + d] + b_hh[2 * DEC_ + d]);
    float og = sigmoidf_(g[3 * DEC_ + d] + b_hh[3 * DEC_ + d]);
    float cn = fg * c[i] + ig * gg;
    float hn = og * tanhf(cn);
    if (t < lens[b] - 1) { h[i] = hn; c[i] = cn; }
}

// ---------------------------------------------------------------------------
extern "C" void kernel_launch(void* const* d_in, const int* in_sizes, int n_in,
                              void* d_out, int out_size, void* d_ws, size_t ws_size,
                              hipStream_t stream)
{
    const float* enc        = (const float*)d_in[0];
    const int*   caps       = (const int*)  d_in[1];
    const int*   lens       = (const int*)  d_in[2];
    const float* table      = (const float*)d_in[3];
    const float* W_enc_att  = (const float*)d_in[4];
    const float* b_enc_att  = (const float*)d_in[5];
    const float* W_dec_att  = (const float*)d_in[6];
    const float* b_dec_att  = (const float*)d_in[7];
    const float* W_full_att = (const float*)d_in[8];
    const float* b_full_att = (const float*)d_in[9];
    const float* W_init_h   = (const float*)d_in[10];
    const float* b_init_h   = (const float*)d_in[11];
    const float* W_init_c   = (const float*)d_in[12];
    const float* b_init_c   = (const float*)d_in[13];
    const float* W_f_beta   = (const float*)d_in[14];
    const float* b_f_beta   = (const float*)d_in[15];
    const float* W_ih       = (const float*)d_in[16];
    const float* b_ih       = (const float*)d_in[17];
    const float* W_hh       = (const float*)d_in[18];
    const float* b_hh       = (const float*)d_in[19];
    const float* W_fc       = (const float*)d_in[20];
    const float* b_fc       = (const float*)d_in[21];
    float* out = (float*)d_out;

    // workspace carve-up (floats): ~34.6 MB total
    float* ws    = (float*)d_ws;
    float* att1  = ws;                               // 25600*256
    float* meanb = att1  + (size_t)B_ * P_ * ATT_;   // 256*1536
    float* h     = meanb + (size_t)B_ * ENC_;        // 256*512
    float* c     = h     + (size_t)B_ * DEC_;
    float* att2  = c     + (size_t)B_ * DEC_;        // 256*256
    float* awe   = att2  + (size_t)B_ * ATT_;        // 256*1536
    float* aweg  = awe   + (size_t)B_ * ENC_;        // 256*1536
    float* gates = aweg  + (size_t)B_ * ENC_;        // 256*2048
    float* embt  = gates + (size_t)B_ * 4 * DEC_;    // 256*256

    const dim3 blk(32, 4); // 4 waves per block, each a 64x16 C tile
    auto gemm = [&](const float* A1, int lda1, const float* W1, int K1,
                    const float* A2, int lda2, const float* W2, int K2,
                    const float* A3, int lda3, const float* W3, int K3,
                    const float* bias, float* C, int M, int N,
                    int mode, const float* aux, int ldaux, const int* L, int t) {
        int tiles_n = (N + 15) / 16;
        dim3 grid((unsigned)((tiles_n + 3) / 4), (unsigned)(M / 64));
        wmma_gemm_f32<<<grid, blk, 0, stream>>>(A1, lda1, W1, K1, A2, lda2, W2, K2,
                                                A3, lda3, W3, K3, bias, C, M, N,
                                                mode, aux, ldaux, L, t);
    };

    // ---- setup ----
    mean_kernel<<<B_, 256, 0, stream>>>(enc, meanb);
    gemm(meanb, ENC_, W_init_h, ENC_, nullptr, 0, nullptr, 0, nullptr, 0, nullptr, 0,
         b_init_h, h, B_, DEC_, 0, nullptr, 0, nullptr, 0);
    gemm(meanb, ENC_, W_init_c, ENC_, nullptr, 0, nullptr, 0, nullptr, 0, nullptr, 0,
         b_init_c, c, B_, DEC_, 0, nullptr, 0, nullptr, 0);
    // att1 = encoder_out @ W_enc_att + b  (25600 x 256, K=1536)
    gemm(enc, ENC_, W_enc_att, ENC_, nullptr, 0, nullptr, 0, nullptr, 0, nullptr, 0,
         b_enc_att, att1, B_ * P_, ATT_, 0, nullptr, 0, nullptr, 0);

    // ---- time loop ----
    for (int t = 0; t < T_; ++t) {
        emb_gather<<<(B_ * EMB_ + 255) / 256, 256, 0, stream>>>(caps, table, embt, t);

        // att2 = h @ W_dec_att + b
        gemm(h, DEC_, W_dec_att, DEC_, nullptr, 0, nullptr, 0, nullptr, 0, nullptr, 0,
             b_dec_att, att2, B_, ATT_, 0, nullptr, 0, nullptr, 0);

        // e / softmax / awe
        attention_kernel<<<B_, 256, 0, stream>>>(att1, att2, W_full_att, b_full_att, enc, awe);

        // aweg = sigmoid(h @ W_f_beta + b) * awe
        gemm(h, DEC_, W_f_beta, DEC_, nullptr, 0, nullptr, 0, nullptr, 0, nullptr, 0,
             b_f_beta, aweg, B_, ENC_, 1, awe, ENC_, nullptr, 0);

        // gates = emb @ W_ih[:EMB] + aweg @ W_ih[EMB:] + h @ W_hh + b_ih
        // (b_hh is added inside lstm_cell)
        gemm(embt, EMB_, W_ih, EMB_,
             aweg, ENC_, W_ih + (size_t)EMB_ * 4 * DEC_, ENC_,
             h,    DEC_, W_hh, DEC_,
             b_ih, gates, B_, 4 * DEC_, 0, nullptr, 0, nullptr, 0);

        lstm_cell<<<(B_ * DEC_ + 255) / 256, 256, 0, stream>>>(gates, b_hh, h, c, lens, t);

        // preds[:, t, :] = mask ? h @ W_fc + b_fc : 0     (out layout B x T x V)
        gemm(h, DEC_, W_fc, DEC_, nullptr, 0, nullptr, 0, nullptr, 0, nullptr, 0,
             b_fc, out + (size_t)t * V_, B_, V_, 2, nullptr, 0, lens, t);
    }
}